// GINLayerModified_1039382086070
// MI455X (gfx1250) — compile-verified
//
#include <hip/hip_runtime.h>
#include <hip/hip_bf16.h>

#define N_NODES  50000
#define N_EDGES  800000
#define EDGE_DIM 16
#define EMB      128
#define HID      256
#define BN_EPS   1e-5f

typedef __attribute__((ext_vector_type(2))) float v2f;
typedef __attribute__((ext_vector_type(8))) float v8f;

// D = A(16x4 f32) * B(4x16 f32) + C(16x16 f32), wave32.
//   A: element (m,k) -> lane = (m&15) + 16*(k>>1), slot = k&1
//   B: element (k,n) -> lane = (n&15) + 16*(k>>1), slot = k&1
//   C/D: element (m,n) -> lane = (n&15) + 16*(m>>3), vgpr r = m&7
__device__ __forceinline__ v8f wmma_f32_16x16x4(v2f a, v2f b, v8f c) {
    return __builtin_amdgcn_wmma_f32_16x16x4_f32(false, a, false, b,
                                                 (short)0, c, false, false);
}

// ---------------------------------------------------------------------------
// K0: zero agg + BN stat accumulators
// ---------------------------------------------------------------------------
__global__ __launch_bounds__(256) void zero_ws_kernel(float4* __restrict__ agg4,
                                                      float4* __restrict__ stats4) {
    int i = blockIdx.x * blockDim.x + threadIdx.x;
    float4 z = make_float4(0.f, 0.f, 0.f, 0.f);
    if (i < N_NODES * EMB / 4) agg4[i] = z;
    if (i < 64) stats4[i] = z;   // sum[128] + sumsq[128]
}

// ---------------------------------------------------------------------------
// K1: edge embedding (WMMA f32) + gather(node_feats[src]) + atomic scatter-sum
//     1 wave == 16 edges. Block = 8 waves = 128 edges.
//     W_edge (8 KB) staged into LDS via CDNA5 async copy (ASYNCcnt path).
// ---------------------------------------------------------------------------
__global__ __launch_bounds__(256) void edge_embed_scatter_kernel(
    const float* __restrict__ node_feats, const float* __restrict__ edge_feats,
    const float* __restrict__ W_edge,     const float* __restrict__ b_edge,
    const int*   __restrict__ src,        const int*   __restrict__ dst,
    float* __restrict__ agg)
{
    __shared__ float sW[EDGE_DIM * EMB];   // 16x128 row-major [k][n], 8 KB
    __shared__ float sB[EMB];

    int tid = threadIdx.x;

    // Async-copy W_edge -> LDS: 256 lanes x 2 x 16B = 8 KB, no VGPR staging.
    {
        unsigned lds0 = (unsigned)(uintptr_t)(&sW[0]) + (unsigned)tid * 16u;
        unsigned gof0 = (unsigned)tid * 16u;
        asm volatile("global_load_async_to_lds_b128 %0, %1, %2"
                     :: "v"(lds0), "v"(gof0), "s"(W_edge) : "memory");
        asm volatile("global_load_async_to_lds_b128 %0, %1, %2"
                     :: "v"(lds0 + 4096u), "v"(gof0 + 4096u), "s"(W_edge)
                     : "memory");
    }
    if (tid < EMB) sB[tid] = b_edge[tid];
    asm volatile("s_wait_asynccnt 0x0" ::: "memory");
    __syncthreads();

    int wave = tid >> 5;
    int lane = tid & 31;
    int n    = lane & 15;      // N (and A-row M) index for this lane
    int hi   = lane >> 4;      // K-pair selector
    int e0   = (blockIdx.x * 8 + wave) * 16;

    // A fragments: 16 edges x 16 edge features, K-chunks of 4
    v2f a[4];
    {
        const float* rowp = edge_feats + (e0 + n) * EDGE_DIM;
#pragma unroll
        for (int j = 0; j < 4; ++j) {
            a[j].x = rowp[4 * j + 2 * hi];
            a[j].y = rowp[4 * j + 2 * hi + 1];
        }
    }

    // 32-bit element offsets of the 8 edge rows this lane's C registers cover
    int soff[8], doff[8];
#pragma unroll
    for (int r = 0; r < 8; ++r) {
        int m = r + 8 * hi;
        soff[r] = src[e0 + m] * EMB;
        doff[r] = dst[e0 + m] * EMB;
    }

#pragma unroll
    for (int t = 0; t < 8; ++t) {              // 8 col-tiles of 16 -> 128 cols
        int col = t * 16 + n;
        float bias = sB[col];
        v8f c = { bias, bias, bias, bias, bias, bias, bias, bias };
#pragma unroll
        for (int j = 0; j < 4; ++j) {          // K = 16 in chunks of 4
            v2f b;
            b.x = sW[(4 * j + 2 * hi)     * EMB + col];
            b.y = sW[(4 * j + 2 * hi + 1) * EMB + col];
            c = wmma_f32_16x16x4(a[j], b, c);
        }
        // Issue all 8 gathers first (multiple loads in flight), then scatter.
        float g[8];
#pragma unroll
        for (int r = 0; r < 8; ++r) g[r] = node_feats[soff[r] + col];
#pragma unroll
        for (int r = 0; r < 8; ++r)
            atomicAdd(&agg[doff[r] + col], c[r] + g[r]);
    }
}

// ---------------------------------------------------------------------------
// K2: fused MLP  h = relu(agg @ W1 + b1) @ W2 + b2  (+ BN stat accumulation)
//     1 block == 16 node rows; hidden tile staged in LDS between GEMMs.
// ---------------------------------------------------------------------------
__global__ __launch_bounds__(256) void mlp_fused_kernel(
    const float* __restrict__ agg,
    const float* __restrict__ W1, const float* __restrict__ b1,
    const float* __restrict__ W2, const float* __restrict__ b2,
    float* __restrict__ h, float* __restrict__ stats)
{
    __shared__ float sH[16 * HID];             // 16 x 256 hidden tile, 16 KB

    int tid  = threadIdx.x;
    int wave = tid >> 5;
    int lane = tid & 31;
    int n    = lane & 15;
    int hi   = lane >> 4;
    int r0   = blockIdx.x * 16;

    // Preload A fragments of the agg tile (16 rows x 128 K) once per wave
    v2f aA[32];
    {
        const float* rowp = agg + (r0 + n) * EMB;
#pragma unroll
        for (int c = 0; c < 32; ++c) {
            aA[c].x = rowp[4 * c + 2 * hi];
            aA[c].y = rowp[4 * c + 2 * hi + 1];
        }
    }

    // GEMM1: each wave produces 2 of the 16 hidden col-tiles
#pragma unroll
    for (int tt = 0; tt < 2; ++tt) {
        int t   = wave * 2 + tt;
        int col = t * 16 + n;
        float bias = b1[col];
        v8f acc = { bias, bias, bias, bias, bias, bias, bias, bias };
#pragma unroll
        for (int c = 0; c < 32; ++c) {         // K = 128
            v2f b;
            b.x = W1[(4 * c + 2 * hi)     * HID + col];
            b.y = W1[(4 * c + 2 * hi + 1) * HID + col];
            acc = wmma_f32_16x16x4(aA[c], b, acc);
        }
#pragma unroll
        for (int r = 0; r < 8; ++r) {
            int m = r + 8 * hi;
            sH[m * HID + col] = fmaxf(acc[r], 0.0f);   // ReLU
        }
    }
    __syncthreads();

    // GEMM2: each wave produces 1 of the 8 output col-tiles; A from LDS
    {
        int col = wave * 16 + n;
        float bias = b2[col];
        v8f acc = { bias, bias, bias, bias, bias, bias, bias, bias };
        const float* srow = sH + n * HID;      // A row m = lane&15
#pragma unroll
        for (int c = 0; c < 64; ++c) {         // K = 256
            v2f a, b;
            a.x = srow[4 * c + 2 * hi];
            a.y = srow[4 * c + 2 * hi + 1];
            b.x = W2[(4 * c + 2 * hi)     * EMB + col];
            b.y = W2[(4 * c + 2 * hi + 1) * EMB + col];
            acc = wmma_f32_16x16x4(a, b, acc);
        }
        float csum = 0.f, csq = 0.f;
#pragma unroll
        for (int r = 0; r < 8; ++r) {
            int m = r + 8 * hi;
            float v = acc[r];
            h[(r0 + m) * EMB + col] = v;
            csum += v;
            csq  += v * v;
        }
        atomicAdd(&stats[col], csum);          // per-feature sum
        atomicAdd(&stats[EMB + col], csq);     // per-feature sum of squares
    }
}

// ---------------------------------------------------------------------------
// K3: fold BN stats into per-column scale/shift
// ---------------------------------------------------------------------------
__global__ void bn_finalize_kernel(const float* __restrict__ gamma,
                                   const float* __restrict__ beta,
                                   float* __restrict__ stats) {
    int col = threadIdx.x;
    float inv_n = 1.0f / (float)N_NODES;
    float mean  = stats[col] * inv_n;
    float var   = stats[EMB + col] * inv_n - mean * mean;   // biased variance
    float sc    = rsqrtf(var + BN_EPS) * gamma[col];
    stats[2 * EMB + col] = sc;
    stats[3 * EMB + col] = beta[col] - mean * sc;
}

// ---------------------------------------------------------------------------
// K4: out = h * scale[col] + shift[col], float4-vectorized
// ---------------------------------------------------------------------------
__global__ __launch_bounds__(256) void bn_apply_kernel(
    const float* __restrict__ h, const float* __restrict__ stats,
    float* __restrict__ out)
{
    int i4   = blockIdx.x * blockDim.x + threadIdx.x;
    int col  = (i4 * 4) & (EMB - 1);           // multiple of 4
    float4 hv = ((const float4*)h)[i4];
    float4 sc = *(const float4*)(stats + 2 * EMB + col);
    float4 sh = *(const float4*)(stats + 3 * EMB + col);
    float4 o;
    o.x = hv.x * sc.x + sh.x;
    o.y = hv.y * sc.y + sh.y;
    o.z = hv.z * sc.z + sh.z;
    o.w = hv.w * sc.w + sh.w;
    ((float4*)out)[i4] = o;
}

// ---------------------------------------------------------------------------
extern "C" void kernel_launch(void* const* d_in, const int* in_sizes, int n_in,
                              void* d_out, int out_size, void* d_ws, size_t ws_size,
                              hipStream_t stream) {
    const float* node_feats = (const float*)d_in[0];
    const float* edge_feats = (const float*)d_in[1];
    const float* W_edge     = (const float*)d_in[2];
    const float* b_edge     = (const float*)d_in[3];
    const float* W1         = (const float*)d_in[4];
    const float* b1         = (const float*)d_in[5];
    const float* W2         = (const float*)d_in[6];
    const float* b2         = (const float*)d_in[7];
    const float* gamma      = (const float*)d_in[8];
    const float* beta       = (const float*)d_in[9];
    const int*   src        = (const int*)d_in[10];
    const int*   dst        = (const int*)d_in[11];
    float* out = (float*)d_out;

    float* ws    = (float*)d_ws;
    float* agg   = ws;                                 // 50000*128 floats
    float* h     = ws + (long)N_NODES * EMB;           // 50000*128 floats
    float* stats = ws + 2L * N_NODES * EMB;            // 512 floats

    zero_ws_kernel<<<(N_NODES * EMB / 4 + 255) / 256, 256, 0, stream>>>(
        (float4*)agg, (float4*)stats);

    edge_embed_scatter_kernel<<<N_EDGES / 16 / 8, 256, 0, stream>>>(
        node_feats, edge_feats, W_edge, b_edge, src, dst, agg);

    mlp_fused_kernel<<<N_NODES / 16, 256, 0, stream>>>(
        agg, W1, b1, W2, b2, h, stats);

    bn_finalize_kernel<<<1, EMB, 0, stream>>>(gamma, beta, stats);

    bn_apply_kernel<<<N_NODES * EMB / 4 / 256, 256, 0, stream>>>(h, stats, out);
}